// MetaGNN_11690900979943
// MI455X (gfx1250) — compile-verified
//
#include <hip/hip_runtime.h>
#include <hip/hip_bf16.h>

typedef __attribute__((ext_vector_type(2))) float    v2f;
typedef __attribute__((ext_vector_type(8))) float    v8f;
typedef __attribute__((ext_vector_type(4))) unsigned v4u;
typedef __attribute__((ext_vector_type(8))) int      v8i;
typedef __attribute__((ext_vector_type(4))) int      v4i;

#define HID   128
#define LDSP  132   // padded stride (floats) for W^T in LDS: bank = (4c + k) % 64
#define LDSA  132   // padded stride for A tile (TDM pad: +4 DWORDs per 128 DWORDs)
#define MROWS 64    // rows per block (4 m-tiles of 16)

// ---------------- utility kernels ----------------

__global__ void fill_f32(float* __restrict__ p, float v, int n) {
    int i = blockIdx.x * blockDim.x + threadIdx.x;
    if (i < n) p[i] = v;
}

__global__ void degree_kernel(const long long* __restrict__ col,
                              const float* __restrict__ w,
                              float* __restrict__ deg, int E) {
    int e = blockIdx.x * blockDim.x + threadIdx.x;
    if (e < E) unsafeAtomicAdd(&deg[col[e]], w[e]);
}

__global__ void dinv_kernel(float* __restrict__ d, int n) {
    int i = blockIdx.x * blockDim.x + threadIdx.x;
    if (i < n) {
        float v = d[i];
        d[i] = (v > 0.f) ? rsqrtf(v) : 0.f;
    }
}

// ---- FP32 WMMA GEMM: out[r][c] = rowscale[r] * sum_k A[r][k]*W[k][c], K=N=128 ----
// Block = 256 thr = 8 waves = 64 rows x 128 cols. Wave wv owns n-tile wv (16 cols),
// holds its 32 B fragments (v2f) in registers, and interleaves 4 m-tile accumulator
// chains per K-step. A tile staged by the Tensor Data Mover with LDS padding
// (stride 132 floats) so every ds_load_b64 A-fetch is bank-conflict-free.
__global__ __launch_bounds__(256) void gemm_wmma(const float* __restrict__ A,
                                                 const float* __restrict__ Wg,
                                                 const float* __restrict__ rowscale,
                                                 float* __restrict__ out, int M) {
    extern __shared__ float smem[];
    float* Wt   = smem;                 // [128][LDSP] transposed: Wt[c*LDSP+k] = W[k][c]
    float* Alds = smem + HID * LDSP;    // [MROWS][LDSA] via TDM pad

    const int m0 = blockIdx.x * MROWS;  // wave-uniform

    // --- TDM: async DMA of the 64x128 fp32 A tile into padded LDS (wave 0 issues) ---
    if ((threadIdx.x >> 5) == 0) {
        unsigned           ldsA = (unsigned)(uintptr_t)(void*)Alds;  // low 32b = LDS offset
        unsigned long long ga   = (unsigned long long)(uintptr_t)(A + (size_t)m0 * HID);
        int tdim1 = M - m0;             // rows available; TDM zero-fills OOB rows
        // D# group 0: count=1 | lds_addr | global_addr[56:0] | type=2
        v4u g0 = { 1u, ldsA, (unsigned)(ga & 0xFFFFFFFFu),
                   (unsigned)(((ga >> 32) & 0x01FFFFFFu) | (2u << 30)) };
        // D# group 1: data_size=4B(2) | pad_enable | pad_interval=128dw(6) | pad_amount=4dw(3)
        //             tensor_dim0=128, tensor_dim1=tdim1, tile_dim0=128, tile_dim1=64,
        //             tensor_dim0_stride=128
        v8i g1 = { (int)((2u << 16) | (1u << 20) | (6u << 22) | (3u << 25)),
                   (int)(128u << 16),                          // tensor_dim0[15:0] @ [63:48]
                   (int)(((unsigned)tdim1 & 0xFFFFu) << 16),   // tensor_dim1[15:0] @ [95:80]
                   (int)((((unsigned)tdim1 >> 16) & 0xFFFFu) | (128u << 16)), // dim1 hi | tile_dim0
                   (int)64,                                    // tile_dim1=64, tile_dim2=0
                   (int)128,                                   // tensor_dim0_stride[31:0]
                   0, 0 };
        v4i gz4 = { 0, 0, 0, 0 };       // 2-D tensor: groups 2/3 NULL
        v8i gz8 = { 0, 0, 0, 0, 0, 0, 0, 0 };
        __builtin_amdgcn_tensor_load_to_lds(g0, g1, gz4, gz4, gz8, 0);
        __builtin_amdgcn_s_wait_tensorcnt(0);
    }

    // --- all 256 threads: stage W transposed into LDS (coalesced global reads) ---
    for (int i = threadIdx.x; i < HID * HID; i += 256) {
        int k = i >> 7, c = i & 127;
        Wt[c * LDSP + k] = Wg[i];
    }
    __syncthreads();

    const int lane  = threadIdx.x & 31;
    const int nt    = threadIdx.x >> 5;      // wave id == n-tile
    const int half  = lane >> 4;             // 0: K=0,1 lanes   1: K=2,3 lanes
    const int khalf = half << 1;
    const int cidx  = lane & 15;

    // B fragments for all 32 K-steps, register-resident (64 VGPRs)
    v2f breg[32];
#pragma unroll
    for (int k = 0; k < 32; ++k)
        breg[k] = *(const v2f*)&Wt[(nt * 16 + cidx) * LDSP + 4 * k + khalf];

    v8f acc[4] = {};
#pragma unroll
    for (int k = 0; k < 32; ++k) {
#pragma unroll
        for (int mt = 0; mt < 4; ++mt) {     // 4 independent WMMA chains per K-step
            v2f a = *(const v2f*)&Alds[(mt * 16 + cidx) * LDSA + 4 * k + khalf];
            acc[mt] = __builtin_amdgcn_wmma_f32_16x16x4_f32(
                false, a, false, breg[k], (short)0, acc[mt], false, false);
        }
    }

    // C/D layout: VGPR v -> row (v + 8*half); col = cidx. Fuse dinv[row] scaling.
#pragma unroll
    for (int mt = 0; mt < 4; ++mt) {
        int base_r = m0 + mt * 16;
        if (base_r >= M) break;              // wave-uniform guard (M % 16 == 0)
#pragma unroll
        for (int v = 0; v < 8; ++v) {
            int r = base_r + v + 8 * half;
            float s = rowscale[r];
            out[(size_t)r * HID + nt * 16 + cidx] = acc[mt][v] * s;
        }
    }
}

// ---------------- edge scatter: agg[col] += w * hs[row] (one wave per edge) ----------------
__global__ void scatter_kernel(const long long* __restrict__ row,
                               const long long* __restrict__ col,
                               const float* __restrict__ w,
                               const float* __restrict__ hs,
                               float* __restrict__ agg, int E) {
    int gid  = blockIdx.x * blockDim.x + threadIdx.x;
    int e    = gid >> 5;
    int lane = gid & 31;
    if (e >= E) return;
    long long r  = row[e];
    long long c  = col[e];
    float     wt = w[e];
    float4 v = ((const float4*)(hs + (size_t)r * HID))[lane];   // 512B coalesced per wave
    float* dst = agg + (size_t)c * HID + (lane << 2);
    unsafeAtomicAdd(dst + 0, v.x * wt);
    unsafeAtomicAdd(dst + 1, v.y * wt);
    unsafeAtomicAdd(dst + 2, v.z * wt);
    unsafeAtomicAdd(dst + 3, v.w * wt);
}

// ---------------- y = dinv[r]*agg[r] + b, fused per-feature sum/sumsq partials ----------------
__global__ __launch_bounds__(128) void finalize_bn(const float* __restrict__ agg,
                                                   const float* __restrict__ dinv,
                                                   const float* __restrict__ bias,
                                                   float* __restrict__ y,
                                                   float* __restrict__ sums,
                                                   float* __restrict__ sumsq, int n) {
    int f = threadIdx.x;                     // one feature per thread
    float b = bias[f];
    float s = 0.f, q = 0.f;
    for (int r = blockIdx.x; r < n; r += gridDim.x) {
        float v = dinv[r] * agg[(size_t)r * HID + f] + b;
        y[(size_t)r * HID + f] = v;
        s += v;
        q += v * v;
    }
    unsafeAtomicAdd(&sums[f], s);
    unsafeAtomicAdd(&sumsq[f], q);
}

__global__ void bn_coef(const float* __restrict__ sums, const float* __restrict__ sumsq,
                        const float* __restrict__ g, const float* __restrict__ be,
                        float* __restrict__ scale, float* __restrict__ shift, float invN) {
    int f = threadIdx.x;
    float mu  = sums[f] * invN;
    float var = sumsq[f] * invN - mu * mu;   // biased variance (training BN)
    float sc  = g[f] * rsqrtf(var + 1e-5f);
    scale[f] = sc;
    shift[f] = be[f] - mu * sc;
}

__global__ void bn_apply(const float* __restrict__ y, const float* __restrict__ scale,
                         const float* __restrict__ shift, float* __restrict__ out,
                         long long n, int relu) {
    long long i = (long long)blockIdx.x * blockDim.x + threadIdx.x;
    if (i >= n) return;
    int f = (int)(i & 127);
    float v = y[i] * scale[f] + shift[f];
    out[i] = relu ? fmaxf(v, 0.f) : v;
}

// ---------------- orchestration ----------------

extern "C" void kernel_launch(void* const* d_in, const int* in_sizes, int n_in,
                              void* d_out, int out_size, void* d_ws, size_t ws_size,
                              hipStream_t stream) {
    const float*     x   = (const float*)d_in[0];
    const long long* ei  = (const long long*)d_in[1];
    const float*     ew  = (const float*)d_in[2];
    const float*     W1  = (const float*)d_in[3];
    const float*     b1  = (const float*)d_in[4];
    const float*     g1  = (const float*)d_in[5];
    const float*     be1 = (const float*)d_in[6];
    const float*     W2  = (const float*)d_in[7];
    const float*     b2  = (const float*)d_in[8];
    const float*     g2  = (const float*)d_in[9];
    const float*     be2 = (const float*)d_in[10];
    float* out = (float*)d_out;

    const int N = in_sizes[0] / HID;
    const int E = in_sizes[2];
    const long long* row = ei;
    const long long* col = ei + E;

    // workspace carve-out
    char* ws = (char*)d_ws;
    size_t off = 0;
    auto carve = [&](size_t bytes) -> float* {
        float* p = (float*)(ws + off);
        off = (off + bytes + 255) & ~(size_t)255;
        return p;
    };
    float* dinv  = carve((size_t)N * 4);
    float* sums  = carve(256 * 4);           // sums[0:128], sumsq[128:256] contiguous
    float* sumsq = sums + 128;
    float* scale = carve(128 * 4);
    float* shift = carve(128 * 4);
    size_t mat   = (size_t)N * HID * 4;
    float* bufA  = carve(mat);
    float* bufB  = carve(mat);
    float* bufC  = carve(mat);

    const int T = 256;
    const int gblocks  = (N + MROWS - 1) / MROWS;
    const size_t shmem = (HID * LDSP + MROWS * LDSA) * sizeof(float);
    const int sblocks  = (int)(((long long)E * 32 + T - 1) / T);
    const long long total = (long long)N * HID;
    const int eblocks  = (int)((total + T - 1) / T);

    // degree + symmetric normalization (deg seeded with 1.0 = self-loop weight)
    fill_f32<<<(N + T - 1) / T, T, 0, stream>>>(dinv, 1.0f, N);
    degree_kernel<<<(E + T - 1) / T, T, 0, stream>>>(col, ew, dinv, E);
    dinv_kernel<<<(N + T - 1) / T, T, 0, stream>>>(dinv, N);

    // ---- layer 1 ----
    gemm_wmma<<<gblocks, 256, shmem, stream>>>(x, W1, dinv, bufA, N);   // hs = (x@W1)*dinv[row]
    (void)hipMemcpyAsync(bufB, bufA, mat, hipMemcpyDeviceToDevice, stream); // agg init = self-loop
    scatter_kernel<<<sblocks, T, 0, stream>>>(row, col, ew, bufA, bufB, E);
    fill_f32<<<1, 256, 0, stream>>>(sums, 0.f, 256);
    finalize_bn<<<1024, 128, 0, stream>>>(bufB, dinv, b1, bufC, sums, sumsq, N);
    bn_coef<<<1, 128, 0, stream>>>(sums, sumsq, g1, be1, scale, shift, 1.0f / (float)N);
    bn_apply<<<eblocks, T, 0, stream>>>(bufC, scale, shift, bufA, total, 1);  // + ReLU

    // ---- layer 2 ----
    gemm_wmma<<<gblocks, 256, shmem, stream>>>(bufA, W2, dinv, bufC, N);
    (void)hipMemcpyAsync(bufB, bufC, mat, hipMemcpyDeviceToDevice, stream);
    scatter_kernel<<<sblocks, T, 0, stream>>>(row, col, ew, bufC, bufB, E);
    fill_f32<<<1, 256, 0, stream>>>(sums, 0.f, 256);
    finalize_bn<<<1024, 128, 0, stream>>>(bufB, dinv, b2, bufA, sums, sumsq, N);
    bn_coef<<<1, 128, 0, stream>>>(sums, sumsq, g2, be2, scale, shift, 1.0f / (float)N);
    bn_apply<<<eblocks, T, 0, stream>>>(bufA, scale, shift, out, total, 0);   // no ReLU
}